// ResidualGatedGATModule_43516608643169
// MI455X (gfx1250) — compile-verified
//
#include <hip/hip_runtime.h>
#include <math.h>

#define NNODES 30000
#define NEDGES 240000
#define NEA    (NEDGES + NNODES)   // edges + self loops = 270000
#define NGRAPH 300
#define NPG    100                 // nodes per graph
#define DIN    128
#define DD     130
#define DPAD   132                 // padded K for GEMM1 / row stride for 130-wide node state
#define DPAD2  144                 // padded N for GEMM2 (9 tiles of 16, 3 groups of 48)
#define NH     4
#define NC     128
#define NHD    512                 // NH * NC
#define NOUT   256

typedef float v2f __attribute__((ext_vector_type(2)));
typedef float v8f __attribute__((ext_vector_type(8)));

// ---------------- helpers ----------------

__device__ __forceinline__ void atomAddF(float* p, float v) {
  unsafeAtomicAdd(p, v);   // lowers to global_atomic_add_f32
}

// monotonic float<->uint encoding for atomicMax on floats
__device__ __forceinline__ unsigned enc_f(float f) {
  unsigned u = __float_as_uint(f);
  return (u & 0x80000000u) ? ~u : (u | 0x80000000u);
}
__device__ __forceinline__ float dec_f(unsigned u) {
  return (u & 0x80000000u) ? __uint_as_float(u & 0x7FFFFFFFu)
                           : __uint_as_float(~u);
}

__global__ void fill_f(float* p, size_t n, float v) {
  size_t i = (size_t)blockIdx.x * blockDim.x + threadIdx.x;
  size_t st = (size_t)gridDim.x * blockDim.x;
  for (; i < n; i += st) p[i] = v;
}

// ---------------- stage x = concat(node_features, tag, density), pad to 132 ----------------

__global__ void build_x(const float* __restrict__ nf, const float* __restrict__ pf,
                        const int* __restrict__ pm, float* __restrict__ x) {
  size_t i = (size_t)blockIdx.x * blockDim.x + threadIdx.x;
  size_t st = (size_t)gridDim.x * blockDim.x;
  size_t tot = (size_t)NNODES * DPAD;
  for (; i < tot; i += st) {
    int n = (int)(i / DPAD), c = (int)(i % DPAD);
    float v;
    if (c < DIN) v = nf[(size_t)n * DIN + c];
    else if (c == DIN) {                      // solvent tag
      bool tag = (n == NNODES - 1) || (pm[n + 1] != pm[n]);
      v = tag ? 1.0f : 0.0f;
    } else if (c == DIN + 1) v = pf[pm[n] * 2 + 0];  // density
    else v = 0.0f;                            // pad cols 130,131
    x[i] = v;
  }
}

// ---------------- zero-padded weight copies (branchless GEMM inner loops) ----------------

__global__ void pad_w1(const float* __restrict__ W1, float* __restrict__ W1p) {
  int i = blockIdx.x * blockDim.x + threadIdx.x;     // [DPAD,512]
  if (i >= DPAD * NHD) return;
  int r = i / NHD, c = i % NHD;
  W1p[i] = (r < DD) ? W1[(size_t)r * NHD + c] : 0.f;
}

__global__ void pad_w2(const float* __restrict__ W2, float* __restrict__ W2p) {
  int i = blockIdx.x * blockDim.x + threadIdx.x;     // [512,DPAD2]
  if (i >= NHD * DPAD2) return;
  int r = i / DPAD2, c = i % DPAD2;
  W2p[i] = (c < DD) ? W2[(size_t)r * DD + c] : 0.f;
}

// ---------------- tiny precomputations ----------------
// stat layout: [0,1]=edge_attr column sums, [2..9]=M1[j*4+h], [10,11]=M2[j]

__global__ void ea_sum_kernel(const float* __restrict__ ea, float* __restrict__ stat) {
  int i = blockIdx.x * blockDim.x + threadIdx.x;
  int st = gridDim.x * blockDim.x;
  float s0 = 0.f, s1 = 0.f;
  for (; i < NEDGES; i += st) { s0 += ea[2 * i]; s1 += ea[2 * i + 1]; }
  atomAddF(&stat[0], s0);
  atomAddF(&stat[1], s1);
}

__global__ void mcoef_kernel(const float* __restrict__ We1, const float* __restrict__ ae1,
                             const float* __restrict__ We2, const float* __restrict__ ae2,
                             float* __restrict__ stat) {
  int t = threadIdx.x;
  if (t < 8) {                       // M1[j,h] = sum_c We1[j, h*128+c] * ae1[h,c]
    int j = t >> 2, h = t & 3;
    float s = 0.f;
    for (int c = 0; c < NC; ++c)
      s += We1[(size_t)j * NHD + h * NC + c] * ae1[h * NC + c];
    stat[2 + j * 4 + h] = s;
  } else if (t < 10) {               // M2[j] = sum_c We2[j,c] * ae2[c]
    int j = t - 8;
    float s = 0.f;
    for (int c = 0; c < DD; ++c) s += We2[(size_t)j * DD + c] * ae2[c];
    stat[10 + j] = s;
  }
}

// ---------------- WMMA GEMM 1: x[N,132] @ W1p[132,512] -> hx[N,512] ----------------
// one wave = 16x64 strip (4 accumulator tiles share each A fragment)

__global__ __launch_bounds__(256)
void gemm1_wmma(const float* __restrict__ X, const float* __restrict__ Wp,
                float* __restrict__ Y) {
  const int TG = NHD / 64;                      // 8 column groups
  int wid = threadIdx.x >> 5, lane = threadIdx.x & 31;
  int tile = blockIdx.x * 8 + wid;              // (N/16)*8 = 15000 exactly
  int tm = tile / TG, tg = tile % TG;
  int mrow  = tm * 16 + (lane & 15);
  int ncol0 = tg * 64 + (lane & 15);
  int kb = (lane >> 4) * 2;
  const float* Xr = X + (size_t)mrow * DPAD + kb;
  v8f a0 = {}, a1 = {}, a2 = {}, a3 = {};
  for (int k0 = 0; k0 < DPAD; k0 += 4) {        // 33 branchless steps
    v2f a; a.x = Xr[k0]; a.y = Xr[k0 + 1];
    const float* Wr0 = Wp + (size_t)(k0 + kb) * NHD + ncol0;
    const float* Wr1 = Wr0 + NHD;
    v2f b0, b1, b2, b3;
    b0.x = Wr0[0];  b0.y = Wr1[0];
    b1.x = Wr0[16]; b1.y = Wr1[16];
    b2.x = Wr0[32]; b2.y = Wr1[32];
    b3.x = Wr0[48]; b3.y = Wr1[48];
    a0 = __builtin_amdgcn_wmma_f32_16x16x4_f32(false, a, false, b0, (short)0, a0, false, false);
    a1 = __builtin_amdgcn_wmma_f32_16x16x4_f32(false, a, false, b1, (short)0, a1, false, false);
    a2 = __builtin_amdgcn_wmma_f32_16x16x4_f32(false, a, false, b2, (short)0, a2, false, false);
    a3 = __builtin_amdgcn_wmma_f32_16x16x4_f32(false, a, false, b3, (short)0, a3, false, false);
  }
  int rbase = tm * 16 + (lane >> 4) * 8;
#pragma unroll
  for (int r = 0; r < 8; ++r) {
    float* yr = Y + (size_t)(rbase + r) * NHD + ncol0;
    yr[0]  = a0[r];
    yr[16] = a1[r];
    yr[32] = a2[r];
    yr[48] = a3[r];
  }
}

// ---------------- WMMA GEMM 2: y[N,512] @ W2p[512,144] -> z[N,130(pad132)] ----------------
// one wave = 16x48 strip (3 accumulator tiles)

__global__ __launch_bounds__(256)
void gemm2_wmma(const float* __restrict__ Yin, const float* __restrict__ Wp,
                float* __restrict__ Z) {
  const int TG = DPAD2 / 48;                    // 3 column groups
  int wid = threadIdx.x >> 5, lane = threadIdx.x & 31;
  int tile = blockIdx.x * 8 + wid;
  if (tile >= (NNODES / 16) * TG) return;       // uniform per wave
  int tm = tile / TG, tg = tile % TG;
  int mrow  = tm * 16 + (lane & 15);
  int ncol0 = tg * 48 + (lane & 15);
  int kb = (lane >> 4) * 2;
  const float* Yr = Yin + (size_t)mrow * NHD + kb;
  v8f a0 = {}, a1 = {}, a2 = {};
#pragma unroll 4
  for (int k0 = 0; k0 < NHD; k0 += 4) {         // 128 branchless steps
    v2f a; a.x = Yr[k0]; a.y = Yr[k0 + 1];
    const float* Wr0 = Wp + (size_t)(k0 + kb) * DPAD2 + ncol0;
    const float* Wr1 = Wr0 + DPAD2;
    v2f b0, b1, b2;
    b0.x = Wr0[0];  b0.y = Wr1[0];
    b1.x = Wr0[16]; b1.y = Wr1[16];
    b2.x = Wr0[32]; b2.y = Wr1[32];
    a0 = __builtin_amdgcn_wmma_f32_16x16x4_f32(false, a, false, b0, (short)0, a0, false, false);
    a1 = __builtin_amdgcn_wmma_f32_16x16x4_f32(false, a, false, b1, (short)0, a1, false, false);
    a2 = __builtin_amdgcn_wmma_f32_16x16x4_f32(false, a, false, b2, (short)0, a2, false, false);
  }
  int rbase = tm * 16 + (lane >> 4) * 8;
#pragma unroll
  for (int r = 0; r < 8; ++r) {
    float* zr = Z + (size_t)(rbase + r) * DPAD;
    int c0 = ncol0, c1 = ncol0 + 16, c2 = ncol0 + 32;
    if (c0 < DD) zr[c0] = a0[r];
    if (c1 < DD) zr[c1] = a1[r];
    if (c2 < DD) zr[c2] = a2[r];
  }
}

// ---------------- attention scalar precomputes ----------------

__global__ void asrc1_kernel(const float* __restrict__ hx, const float* __restrict__ as,
                             const float* __restrict__ ad, float* __restrict__ osrc,
                             float* __restrict__ odst) {
  int idx = blockIdx.x * blockDim.x + threadIdx.x;
  if (idx >= NNODES * NH) return;
  int n = idx >> 2, h = idx & 3;
  const float* hr = hx + (size_t)n * NHD + h * NC;
  float s1 = 0.f, s2 = 0.f;
  for (int c = 0; c < NC; ++c) {
    float v = hr[c];
    s1 += v * as[h * NC + c];
    s2 += v * ad[h * NC + c];
  }
  osrc[idx] = s1; odst[idx] = s2;
}

__global__ void asrc2_kernel(const float* __restrict__ z, const float* __restrict__ as,
                             const float* __restrict__ ad, float* __restrict__ osrc,
                             float* __restrict__ odst) {
  int n = blockIdx.x * blockDim.x + threadIdx.x;
  if (n >= NNODES) return;
  const float* zr = z + (size_t)n * DPAD;
  float s1 = 0.f, s2 = 0.f;
  for (int c = 0; c < DD; ++c) { s1 += zr[c] * as[c]; s2 += zr[c] * ad[c]; }
  osrc[n] = s1; odst[n] = s2;
}

// ---------------- edge softmax passes (conv1, H=4) ----------------

__global__ void edge1_logits(const int* __restrict__ ei, const float* __restrict__ ea,
                             const float* __restrict__ stat,
                             const float* __restrict__ asrc, const float* __restrict__ adst,
                             float* __restrict__ ex, unsigned* __restrict__ menc) {
  const float invE = 1.0f / (float)NEDGES;
  int e = blockIdx.x * blockDim.x + threadIdx.x;
  int st = gridDim.x * blockDim.x;
  for (; e < NEA; e += st) {
    int s, d; float e0, e1;
    if (e < NEDGES) { s = ei[e]; d = ei[NEDGES + e]; e0 = ea[2 * e]; e1 = ea[2 * e + 1]; }
    else { s = d = e - NEDGES; e0 = stat[0] * invE; e1 = stat[1] * invE; }
#pragma unroll
    for (int h = 0; h < NH; ++h) {
      float al = asrc[s * NH + h] + adst[d * NH + h]
               + e0 * stat[2 + h] + e1 * stat[6 + h];
      al = (al > 0.f) ? al : 0.2f * al;        // leaky relu
      ex[(size_t)e * NH + h] = al;             // store logit
      atomicMax(&menc[d * NH + h], enc_f(al));
    }
  }
}

__global__ void edge1_norm(const int* __restrict__ ei, const unsigned* __restrict__ menc,
                           float* __restrict__ ex, float* __restrict__ den) {
  int e = blockIdx.x * blockDim.x + threadIdx.x;
  int st = gridDim.x * blockDim.x;
  for (; e < NEA; e += st) {
    int d = (e < NEDGES) ? ei[NEDGES + e] : (e - NEDGES);
#pragma unroll
    for (int h = 0; h < NH; ++h) {
      float v = expf(ex[(size_t)e * NH + h] - dec_f(menc[d * NH + h]));
      ex[(size_t)e * NH + h] = v;
      atomAddF(&den[d * NH + h], v);
    }
  }
}

__global__ __launch_bounds__(256)
void agg1_kernel(const int* __restrict__ ei, const float* __restrict__ hx,
                 const float* __restrict__ ex, const float* __restrict__ den,
                 float* __restrict__ out) {
  int wid = threadIdx.x >> 5, lane = threadIdx.x & 31;
  int e = blockIdx.x * 8 + wid;
  if (e >= NEA) return;
  int s, d;
  if (e < NEDGES) { s = ei[e]; d = ei[NEDGES + e]; } else { s = d = e - NEDGES; }
  float w[NH];
#pragma unroll
  for (int h = 0; h < NH; ++h)
    w[h] = ex[(size_t)e * NH + h] / den[d * NH + h];
  const float* hs = hx + (size_t)s * NHD;
  float* od = out + (size_t)d * NHD;
  for (int c = lane; c < NHD; c += 32)
    atomAddF(&od[c], hs[c] * w[c >> 7]);
}

// ---------------- bias + LN + relu over 512 ----------------

__global__ __launch_bounds__(256)
void ln1_kernel(const float* __restrict__ agg, const float* __restrict__ b,
                const float* __restrict__ g, const float* __restrict__ beta,
                float* __restrict__ y) {
  __shared__ float ss[256], sq[256];
  int n = blockIdx.x, t = threadIdx.x;
  const float* a = agg + (size_t)n * NHD;
  float v0 = a[t] + b[t], v1 = a[t + 256] + b[t + 256];
  ss[t] = v0 + v1; sq[t] = v0 * v0 + v1 * v1;
  __syncthreads();
  for (int st = 128; st > 0; st >>= 1) {
    if (t < st) { ss[t] += ss[t + st]; sq[t] += sq[t + st]; }
    __syncthreads();
  }
  float mu = ss[0] * (1.f / NHD);
  float var = sq[0] * (1.f / NHD) - mu * mu;
  float rs = rsqrtf(var + 1e-5f);
  float o0 = (v0 - mu) * rs * g[t] + beta[t];
  float o1 = (v1 - mu) * rs * g[t + 256] + beta[t + 256];
  y[(size_t)n * NHD + t]       = o0 > 0.f ? o0 : 0.f;
  y[(size_t)n * NHD + t + 256] = o1 > 0.f ? o1 : 0.f;
}

// ---------------- edge softmax passes (conv2, H=1) ----------------

__global__ void edge2_logits(const int* __restrict__ ei, const float* __restrict__ ea,
                             const float* __restrict__ stat,
                             const float* __restrict__ asrc, const float* __restrict__ adst,
                             float* __restrict__ ex, unsigned* __restrict__ menc) {
  const float invE = 1.0f / (float)NEDGES;
  int e = blockIdx.x * blockDim.x + threadIdx.x;
  int st = gridDim.x * blockDim.x;
  for (; e < NEA; e += st) {
    int s, d; float e0, e1;
    if (e < NEDGES) { s = ei[e]; d = ei[NEDGES + e]; e0 = ea[2 * e]; e1 = ea[2 * e + 1]; }
    else { s = d = e - NEDGES; e0 = stat[0] * invE; e1 = stat[1] * invE; }
    float al = asrc[s] + adst[d] + e0 * stat[10] + e1 * stat[11];
    al = (al > 0.f) ? al : 0.2f * al;
    ex[e] = al;
    atomicMax(&menc[d], enc_f(al));
  }
}

__global__ void edge2_norm(const int* __restrict__ ei, const unsigned* __restrict__ menc,
                           float* __restrict__ ex, float* __restrict__ den) {
  int e = blockIdx.x * blockDim.x + threadIdx.x;
  int st = gridDim.x * blockDim.x;
  for (; e < NEA; e += st) {
    int d = (e < NEDGES) ? ei[NEDGES + e] : (e - NEDGES);
    float v = expf(ex[e] - dec_f(menc[d]));
    ex[e] = v;
    atomAddF(&den[d], v);
  }
}

__global__ __launch_bounds__(256)
void agg2_kernel(const int* __restrict__ ei, const float* __restrict__ z,
                 const float* __restrict__ ex, const float* __restrict__ den,
                 float* __restrict__ out) {
  int wid = threadIdx.x >> 5, lane = threadIdx.x & 31;
  int e = blockIdx.x * 8 + wid;
  if (e >= NEA) return;
  int s, d;
  if (e < NEDGES) { s = ei[e]; d = ei[NEDGES + e]; } else { s = d = e - NEDGES; }
  float w = ex[e] / den[d];
  const float* zs = z + (size_t)s * DPAD;
  float* od = out + (size_t)d * DPAD;
  for (int c = lane; c < DD; c += 32)
    atomAddF(&od[c], zs[c] * w);
}

// ---------------- bias + LN(130) + gated residual, in place ----------------

__global__ __launch_bounds__(128)
void post_kernel(const float* __restrict__ xres, const float* __restrict__ b2,
                 const float* __restrict__ g2, const float* __restrict__ beta2,
                 const float* __restrict__ gw, const float* __restrict__ gb,
                 float* __restrict__ xf) {
  __shared__ float ss[128], sq[128], sg[128];
  int n = blockIdx.x, t = threadIdx.x;
  float* a = xf + (size_t)n * DPAD;
  const float* xr = xres + (size_t)n * DPAD;
  float v0 = a[t] + b2[t];
  float lsum = v0, lsq = v0 * v0, lg = xr[t] * gw[t];
  int c1 = t + 128;
  float v1 = 0.f;
  bool has1 = (c1 < DD);
  if (has1) {
    v1 = a[c1] + b2[c1];
    lsum += v1; lsq += v1 * v1; lg += xr[c1] * gw[c1];
  }
  ss[t] = lsum; sq[t] = lsq; sg[t] = lg;
  __syncthreads();
  for (int st = 64; st > 0; st >>= 1) {
    if (t < st) { ss[t] += ss[t + st]; sq[t] += sq[t + st]; sg[t] += sg[t + st]; }
    __syncthreads();
  }
  float mu = ss[0] * (1.f / DD);
  float var = sq[0] * (1.f / DD) - mu * mu;
  float rs = rsqrtf(var + 1e-5f);
  float gate = 1.f / (1.f + expf(-(sg[0] + gb[0])));
  float o0 = (v0 - mu) * rs * g2[t] + beta2[t];
  a[t] = gate * o0 + (1.f - gate) * xr[t];
  if (has1) {
    float o1 = (v1 - mu) * rs * g2[c1] + beta2[c1];
    a[c1] = gate * o1 + (1.f - gate) * xr[c1];
  }
}

// ---------------- per-graph gated attention pooling ----------------

__global__ __launch_bounds__(128)
void pool_kernel(const float* __restrict__ xf, const float* __restrict__ pw,
                 const float* __restrict__ pb, float* __restrict__ pooled) {
  __shared__ float s_sh[NPG];
  __shared__ float w_sh[NPG];
  __shared__ float dsum;
  int g = blockIdx.x, t = threadIdx.x;
  size_t base = (size_t)g * NPG;
  if (t < NPG) {
    const float* xr = xf + (base + t) * DPAD;
    float acc = 0.f;
    for (int c = 0; c < DD; ++c) acc += xr[c] * pw[c];
    s_sh[t] = 1.f / (1.f + expf(-(acc + pb[0])));
  }
  __syncthreads();
  if (t == 0) {
    float m = -1e30f;
    for (int i = 0; i < NPG - 1; ++i) m = fmaxf(m, s_sh[i]);
    float den = 0.f;
    for (int i = 0; i < NPG - 1; ++i) {
      float v = expf(s_sh[i] - m);
      w_sh[i] = v; den += v;
    }
    dsum = den;
  }
  __syncthreads();
  float rden = 1.f / dsum;
  const float* sol = xf + (base + NPG - 1) * DPAD;   // solvent = last node, weight 1
  for (int c = t; c < DD; c += 128) {
    float acc = 0.f;
    for (int i = 0; i < NPG - 1; ++i) acc += w_sh[i] * xf[(base + i) * DPAD + c];
    pooled[(size_t)g * (2 * DD) + c]      = acc * rden;
    pooled[(size_t)g * (2 * DD) + DD + c] = sol[c];
  }
}

// ---------------- final projection [G,260] @ [260,256] + bias ----------------

__global__ void proj_kernel(const float* __restrict__ pooled, const float* __restrict__ pwm,
                            const float* __restrict__ pbv, float* __restrict__ out) {
  int idx = blockIdx.x * blockDim.x + threadIdx.x;
  if (idx >= NGRAPH * NOUT) return;
  int g = idx / NOUT, o = idx % NOUT;
  float acc = pbv[o];
  const float* pr = pooled + (size_t)g * (2 * DD);
  for (int k = 0; k < 2 * DD; ++k) acc += pr[k] * pwm[(size_t)k * NOUT + o];
  out[idx] = acc;
}

// ---------------- launch ----------------

extern "C" void kernel_launch(void* const* d_in, const int* in_sizes, int n_in,
                              void* d_out, int out_size, void* d_ws, size_t ws_size,
                              hipStream_t stream) {
  const float* nf   = (const float*)d_in[0];
  const float* ea   = (const float*)d_in[1];
  const float* pf   = (const float*)d_in[2];
  const int*   ei   = (const int*)  d_in[3];
  const int*   pm   = (const int*)  d_in[4];
  const float* W1   = (const float*)d_in[5];
  const float* as1  = (const float*)d_in[6];
  const float* ad1  = (const float*)d_in[7];
  const float* We1  = (const float*)d_in[8];
  const float* ae1  = (const float*)d_in[9];
  const float* b1   = (const float*)d_in[10];
  const float* g1   = (const float*)d_in[11];
  const float* bt1  = (const float*)d_in[12];
  const float* W2   = (const float*)d_in[13];
  const float* as2  = (const float*)d_in[14];
  const float* ad2  = (const float*)d_in[15];
  const float* We2  = (const float*)d_in[16];
  const float* ae2  = (const float*)d_in[17];
  const float* b2   = (const float*)d_in[18];
  const float* g2   = (const float*)d_in[19];
  const float* bt2  = (const float*)d_in[20];
  const float* gw   = (const float*)d_in[21];
  const float* gb   = (const float*)d_in[22];
  const float* pw   = (const float*)d_in[23];
  const float* pb   = (const float*)d_in[24];
  const float* pjw  = (const float*)d_in[25];
  const float* pjb  = (const float*)d_in[26];
  float* out = (float*)d_out;

  float* W = (float*)d_ws;
  size_t off = 0;
  float* x_res = W + off; off += (size_t)NNODES * DPAD;
  float* hx    = W + off; off += (size_t)NNODES * NHD;   // reused as y after LN1
  float* agg1  = W + off; off += (size_t)NNODES * NHD;
  float* z     = W + off; off += (size_t)NNODES * DPAD;
  float* agg2  = W + off; off += (size_t)NNODES * DPAD;  // becomes x_final in place
  float* W1p   = W + off; off += (size_t)DPAD * NHD;     // zero-padded W1 [132,512]
  float* W2p   = W + off; off += (size_t)NHD * DPAD2;    // zero-padded W2 [512,144]
  float* asrc1 = W + off; off += (size_t)NNODES * NH;
  float* adst1 = W + off; off += (size_t)NNODES * NH;
  float* den1  = W + off; off += (size_t)NNODES * NH;
  unsigned* menc1 = (unsigned*)(W + off); off += (size_t)NNODES * NH;
  float* ex1   = W + off; off += (size_t)NEA * NH;
  float* asrc2 = W + off; off += NNODES;
  float* adst2 = W + off; off += NNODES;
  float* den2  = W + off; off += NNODES;
  unsigned* menc2 = (unsigned*)(W + off); off += NNODES;
  float* ex2   = W + off; off += NEA;
  float* stat  = W + off; off += 16;
  float* pooled= W + off; off += (size_t)NGRAPH * 2 * DD;
  (void)ws_size; (void)n_in; (void)in_sizes; (void)out_size;

  // zero accumulators (fresh every call)
  fill_f<<<2048, 256, 0, stream>>>(agg1, (size_t)NNODES * NHD, 0.f);
  fill_f<<<512,  256, 0, stream>>>(agg2, (size_t)NNODES * DPAD, 0.f);
  fill_f<<<256,  256, 0, stream>>>(den1, (size_t)NNODES * NH, 0.f);
  fill_f<<<256,  256, 0, stream>>>((float*)menc1, (size_t)NNODES * NH, 0.f);
  fill_f<<<128,  256, 0, stream>>>(den2, NNODES, 0.f);
  fill_f<<<128,  256, 0, stream>>>((float*)menc2, NNODES, 0.f);
  fill_f<<<1,    32,  0, stream>>>(stat, 16, 0.f);

  build_x<<<4096, 256, 0, stream>>>(nf, pf, pm, x_res);
  pad_w1<<<(DPAD * NHD + 255) / 256, 256, 0, stream>>>(W1, W1p);
  pad_w2<<<(NHD * DPAD2 + 255) / 256, 256, 0, stream>>>(W2, W2p);
  ea_sum_kernel<<<512, 256, 0, stream>>>(ea, stat);
  mcoef_kernel<<<1, 32, 0, stream>>>(We1, ae1, We2, ae2, stat);

  // conv1
  gemm1_wmma<<<NNODES / 16, 256, 0, stream>>>(x_res, W1p, hx);   // 1875 blocks * 8 waves
  asrc1_kernel<<<(NNODES * NH + 255) / 256, 256, 0, stream>>>(hx, as1, ad1, asrc1, adst1);
  edge1_logits<<<2048, 256, 0, stream>>>(ei, ea, stat, asrc1, adst1, ex1, menc1);
  edge1_norm<<<2048, 256, 0, stream>>>(ei, menc1, ex1, den1);
  agg1_kernel<<<(NEA + 7) / 8, 256, 0, stream>>>(ei, hx, ex1, den1, agg1);
  ln1_kernel<<<NNODES, 256, 0, stream>>>(agg1, b1, g1, bt1, hx);  // y := hx buffer

  // conv2
  gemm2_wmma<<<((NNODES / 16) * 3 + 7) / 8, 256, 0, stream>>>(hx, W2p, z);
  asrc2_kernel<<<(NNODES + 255) / 256, 256, 0, stream>>>(z, as2, ad2, asrc2, adst2);
  edge2_logits<<<2048, 256, 0, stream>>>(ei, ea, stat, asrc2, adst2, ex2, menc2);
  edge2_norm<<<2048, 256, 0, stream>>>(ei, menc2, ex2, den2);
  agg2_kernel<<<(NEA + 7) / 8, 256, 0, stream>>>(ei, z, ex2, den2, agg2);
  post_kernel<<<NNODES, 128, 0, stream>>>(x_res, b2, g2, bt2, gw, gb, agg2);

  // pooling + projection
  pool_kernel<<<NGRAPH, 128, 0, stream>>>(agg2, pw, pb, pooled);
  proj_kernel<<<(NGRAPH * NOUT + 255) / 256, 256, 0, stream>>>(pooled, pjw, pjb, out);
}